// GNN_53386443489659
// MI455X (gfx1250) — compile-verified
//
#include <hip/hip_runtime.h>

#define N_NODES  50000
#define N_EDGES  800000
#define N_GRAPHS 128
#define F_INDIM  16
#define HDIM     64
#define L_CONV   63

typedef __attribute__((ext_vector_type(2))) float v2f;
typedef __attribute__((ext_vector_type(8))) float v8f;

// ---------------------------------------------------------------- utilities
__global__ void zero_f32(float* __restrict__ p, int n) {
    int i = blockIdx.x * blockDim.x + threadIdx.x;
    if (i < n) p[i] = 0.0f;
}

// one lane per (edge, feature): agg[dst[e]][f] += feat[src[e]][f]
__global__ void scatter_add_feat(const float* __restrict__ feat,
                                 const int* __restrict__ srcIdx,
                                 const int* __restrict__ dstIdx,
                                 float* __restrict__ agg,
                                 int total, int shiftF) {
    int i = blockIdx.x * blockDim.x + threadIdx.x;
    if (i >= total) return;
    int e = i >> shiftF;
    int f = i & ((1 << shiftF) - 1);
    int s = srcIdx[e];
    int d = dstIdx[e];
    unsafeAtomicAdd(&agg[((long)d << shiftF) + f], feat[((long)s << shiftF) + f]);
}

// ------------------------------------------------- dual GEMM via f32 WMMA
// OUT[N,64] = A1[N,K] @ W1[K,64] + A2[N,K] @ W2[K,64] + bias  (opt. ReLU)
// One wave32 computes one 16x16 output tile with V_WMMA_F32_16X16X4_F32.
template <int K, bool RELU>
__global__ void gemm_dual_wmma(const float* __restrict__ A1,
                               const float* __restrict__ W1,
                               const float* __restrict__ A2,
                               const float* __restrict__ W2,
                               const float* __restrict__ bias,
                               float* __restrict__ out,
                               int nNodes) {
    const int lane = threadIdx.x & 31;
    const int waveInBlock = threadIdx.x >> 5;
    const int tile = blockIdx.x * (blockDim.x >> 5) + waveInBlock;
    const int nTilesN = HDIM / 16;                 // 4
    const int nTiles = (nNodes / 16) * nTilesN;
    if (tile >= nTiles) return;                    // wave-uniform exit

    const int mTile = tile / nTilesN;
    const int nTile = tile % nTilesN;

    // A fragment: lanes 0-15 carry K=k+0,k+1; lanes 16-31 carry K=k+2,k+3 (same rows)
    const int aRow  = mTile * 16 + (lane & 15);
    // B fragment: lanes 0-15 = rows K=k+0,k+1 of cols; lanes 16-31 = K=k+2,k+3
    const int col   = nTile * 16 + (lane & 15);
    const int khalf = (lane >> 4) * 2;             // 0 or 2

    v8f acc = {};
    #pragma unroll
    for (int k = 0; k < K; k += 4) {
        v2f a, b;
        a.x = A1[aRow * K + k + khalf];
        a.y = A1[aRow * K + k + khalf + 1];
        b.x = W1[(k + khalf) * HDIM + col];
        b.y = W1[(k + khalf + 1) * HDIM + col];
        acc = __builtin_amdgcn_wmma_f32_16x16x4_f32(false, a, false, b,
                                                    (short)0, acc, false, false);
    }
    #pragma unroll
    for (int k = 0; k < K; k += 4) {
        v2f a, b;
        a.x = A2[aRow * K + k + khalf];
        a.y = A2[aRow * K + k + khalf + 1];
        b.x = W2[(k + khalf) * HDIM + col];
        b.y = W2[(k + khalf + 1) * HDIM + col];
        acc = __builtin_amdgcn_wmma_f32_16x16x4_f32(false, a, false, b,
                                                    (short)0, acc, false, false);
    }

    // C/D layout: VGPR r -> row r (lanes 0-15) / row r+8 (lanes 16-31)
    const float bv = bias[col];
    const int rbase = mTile * 16 + ((lane >> 4) << 3);
    #pragma unroll
    for (int r = 0; r < 8; ++r) {
        float v = acc[r] + bv;
        if (RELU) v = fmaxf(v, 0.0f);
        out[(rbase + r) * HDIM + col] = v;
    }
}

// ---------------------------------- conv1d(1,1,2)^63 chain + mean-pool accum
__global__ void conv_chain_pool(const float* __restrict__ h,
                                const int* __restrict__ batch,
                                const float* __restrict__ cw,   // [L,2]
                                const float* __restrict__ cb,   // [L]
                                float* __restrict__ gsum,
                                float* __restrict__ gcnt,
                                int nNodes) {
    int n = blockIdx.x * blockDim.x + threadIdx.x;
    if (n >= nNodes) return;

    float v[HDIM];
    const float4* hp = (const float4*)(h + (long)n * HDIM);
    #pragma unroll
    for (int i = 0; i < HDIM / 4; ++i) {
        float4 t = hp[i];
        v[4 * i + 0] = t.x; v[4 * i + 1] = t.y;
        v[4 * i + 2] = t.z; v[4 * i + 3] = t.w;
    }
    #pragma unroll
    for (int l = 0; l < L_CONV; ++l) {
        const float w0 = cw[2 * l], w1 = cw[2 * l + 1], b = cb[l];
        #pragma unroll
        for (int j = 0; j < L_CONV - l; ++j)
            v[j] = fmaf(v[j], w0, fmaf(v[j + 1], w1, b));
    }
    int g = batch[n];
    unsafeAtomicAdd(&gsum[g], v[0]);
    unsafeAtomicAdd(&gcnt[g], 1.0f);
}

__global__ void finalize_pool(const float* __restrict__ gsum,
                              const float* __restrict__ gcnt,
                              float* __restrict__ out) {
    int g = threadIdx.x;
    if (g < N_GRAPHS) out[g] = gsum[g] / fmaxf(gcnt[g], 1.0f);
}

// -------------------------------------------------------------------- launch
extern "C" void kernel_launch(void* const* d_in, const int* in_sizes, int n_in,
                              void* d_out, int out_size, void* d_ws, size_t ws_size,
                              hipStream_t stream) {
    const float* x      = (const float*)d_in[0];
    const int*   eidx   = (const int*)d_in[1];
    const int*   srcI   = eidx;             // edge_index[0]
    const int*   dstI   = eidx + N_EDGES;   // edge_index[1]
    const int*   batch  = (const int*)d_in[2];
    const float* Wrel1  = (const float*)d_in[3];
    const float* brel1  = (const float*)d_in[4];
    const float* Wroot1 = (const float*)d_in[5];
    const float* Wrel2  = (const float*)d_in[6];
    const float* brel2  = (const float*)d_in[7];
    const float* Wroot2 = (const float*)d_in[8];
    const float* Wrel3  = (const float*)d_in[9];
    const float* brel3  = (const float*)d_in[10];
    const float* Wroot3 = (const float*)d_in[11];
    const float* convW  = (const float*)d_in[12];
    const float* convB  = (const float*)d_in[13];
    float* out = (float*)d_out;

    float* ws   = (float*)d_ws;
    float* agg  = ws;                                   // [N,64] (layer1 uses [N,16])
    float* hA   = agg + (size_t)N_NODES * HDIM;         // [N,64]
    float* hB   = hA  + (size_t)N_NODES * HDIM;         // [N,64]
    float* gsum = hB  + (size_t)N_NODES * HDIM;         // [G]
    float* gcnt = gsum + N_GRAPHS;                      // [G]

    const int nTiles = (N_NODES / 16) * (HDIM / 16);    // 12500
    const int gemmBlocks = (nTiles + 3) / 4;            // 4 waves (128 thr) per block

    // ---- GraphConv 1: agg = scatter(x); hA = relu(agg@Wrel1 + b1 + x@Wroot1)
    zero_f32<<<(N_NODES * F_INDIM + 255) / 256, 256, 0, stream>>>(agg, N_NODES * F_INDIM);
    scatter_add_feat<<<(N_EDGES * F_INDIM + 255) / 256, 256, 0, stream>>>(
        x, srcI, dstI, agg, N_EDGES * F_INDIM, 4);
    gemm_dual_wmma<F_INDIM, true><<<gemmBlocks, 128, 0, stream>>>(
        agg, Wrel1, x, Wroot1, brel1, hA, N_NODES);

    // ---- GraphConv 2
    zero_f32<<<(N_NODES * HDIM + 255) / 256, 256, 0, stream>>>(agg, N_NODES * HDIM);
    scatter_add_feat<<<(N_EDGES * HDIM + 255) / 256, 256, 0, stream>>>(
        hA, srcI, dstI, agg, N_EDGES * HDIM, 6);
    gemm_dual_wmma<HDIM, true><<<gemmBlocks, 128, 0, stream>>>(
        agg, Wrel2, hA, Wroot2, brel2, hB, N_NODES);

    // ---- GraphConv 3 (no relu)
    zero_f32<<<(N_NODES * HDIM + 255) / 256, 256, 0, stream>>>(agg, N_NODES * HDIM);
    scatter_add_feat<<<(N_EDGES * HDIM + 255) / 256, 256, 0, stream>>>(
        hB, srcI, dstI, agg, N_EDGES * HDIM, 6);
    gemm_dual_wmma<HDIM, false><<<gemmBlocks, 128, 0, stream>>>(
        agg, Wrel3, hB, Wroot3, brel3, hA, N_NODES);

    // ---- conv1d chain + mean pool
    zero_f32<<<1, 256, 0, stream>>>(gsum, 2 * N_GRAPHS);  // gsum+gcnt contiguous
    conv_chain_pool<<<(N_NODES + 255) / 256, 256, 0, stream>>>(
        hA, batch, convW, convB, gsum, gcnt, N_NODES);
    finalize_pool<<<1, 128, 0, stream>>>(gsum, gcnt, out);
}